// AutoEncoder_77137612636645
// MI455X (gfx1250) — compile-verified
//
#include <hip/hip_runtime.h>

// Problem constants (match reference)
#define B_   4096
#define T_   1024
#define F_   8
#define H_   4

#define WAVES_PER_BLOCK 8
#define SAMPLES_PER_WAVE 16
#define PF_DIST 16   // prefetch distance in timesteps (16 * 32B = 512B per lane)

typedef float v2f __attribute__((ext_vector_type(2)));
typedef float v4f __attribute__((ext_vector_type(4)));
typedef float v8f __attribute__((ext_vector_type(8)));

union AccU { v8f v; float f[8]; };

// D = A(16x4 f32) * B(4x16 f32) + C(16x16 f32), one wave.
__device__ __forceinline__ v8f wmma4(v2f a, v2f b, v8f c) {
  return __builtin_amdgcn_wmma_f32_16x16x4_f32(
      /*neg_a=*/false, a, /*neg_b=*/false, b,
      /*c_mod=*/(short)0, c, /*reuse_a=*/false, /*reuse_b=*/false);
}

// Wave-local LDS ordering: DS ops from one wave are in-order; just drain DScnt.
__device__ __forceinline__ void wave_sync_lds() {
  asm volatile("s_wait_dscnt 0" ::: "memory");
}

__device__ __forceinline__ float sigm(float x) {
  return __builtin_amdgcn_rcpf(1.0f + __expf(-x));
}
__device__ __forceinline__ float tanh_f(float x) {
  // tanh(x) = 2*sigmoid(2x) - 1
  return 2.0f * __builtin_amdgcn_rcpf(1.0f + __expf(-2.0f * x)) - 1.0f;
}

// Scatter D-layout accumulator (16 samples x 16 gates) into LDS zbuf[m*16+n].
__device__ __forceinline__ void store_D(float* zb, const AccU& z, int lane, int n) {
#pragma unroll
  for (int v = 0; v < 8; ++v) {
    const int m = (lane < 16) ? v : (v + 8);
    zb[m * 16 + n] = z.f[v];
  }
}

// A-matrix fragment (16x4) from a [16 x 4] LDS buffer.
__device__ __forceinline__ v2f loadA4(const float* hb, int mA, int klo) {
  v2f a; a.x = hb[mA * 4 + klo]; a.y = hb[mA * 4 + klo + 1]; return a;
}
// A-matrix fragment from a [16 x 8] buffer, K-slab khalf in {0,4}.
__device__ __forceinline__ v2f loadA8(const float* xb, int mA, int klo, int khalf) {
  v2f a; a.x = xb[mA * 8 + khalf + klo]; a.y = xb[mA * 8 + khalf + klo + 1]; return a;
}

// Elementwise LSTM update; each lane owns 2 (sample, hidden) pairs.
__device__ __forceinline__ void lstm_ew(const float* zb, float* hb, float c[2],
                                        int m0, int m1, int j) {
  const int mm[2] = { m0, m1 };
#pragma unroll
  for (int p = 0; p < 2; ++p) {
    const int m = mm[p];
    const float gi = zb[m * 16 + j];
    const float gf = zb[m * 16 + 4 + j];
    const float gg = zb[m * 16 + 8 + j];
    const float go = zb[m * 16 + 12 + j];
    const float cn = sigm(gf) * c[p] + sigm(gi) * tanh_f(gg);
    c[p] = cn;
    hb[m * 4 + j] = sigm(go) * tanh_f(cn);
  }
}

// Variant that also writes relu(h) for the decoder FC input.
__device__ __forceinline__ void lstm_ew_relu(const float* zb, float* hb, float* rb,
                                             float c[2], int m0, int m1, int j) {
  const int mm[2] = { m0, m1 };
#pragma unroll
  for (int p = 0; p < 2; ++p) {
    const int m = mm[p];
    const float gi = zb[m * 16 + j];
    const float gf = zb[m * 16 + 4 + j];
    const float gg = zb[m * 16 + 8 + j];
    const float go = zb[m * 16 + 12 + j];
    const float cn = sigm(gf) * c[p] + sigm(gi) * tanh_f(gg);
    c[p] = cn;
    const float h = sigm(go) * tanh_f(cn);
    hb[m * 4 + j] = h;
    rb[m * 4 + j] = fmaxf(h, 0.0f);
  }
}

__global__ __launch_bounds__(256)
void lstm_autoencoder_kernel(
    const float* __restrict__ X,
    const float* __restrict__ eW0, const float* __restrict__ eU0, const float* __restrict__ eb0,
    const float* __restrict__ eW1, const float* __restrict__ eU1, const float* __restrict__ eb1,
    const float* __restrict__ dW0, const float* __restrict__ dU0, const float* __restrict__ db0,
    const float* __restrict__ dW1, const float* __restrict__ dU1, const float* __restrict__ db1,
    const float* __restrict__ fcW, const float* __restrict__ fcb,
    float* __restrict__ out)
{
  const int lane = threadIdx.x & 31;
  const int wave = threadIdx.x >> 5;
  const int gw   = blockIdx.x * WAVES_PER_BLOCK + wave;
  const int s0   = gw * SAMPLES_PER_WAVE;      // base batch sample for this wave

  // Per-wave private LDS slice: z(16x16), h0(16x4), h1(16x4), r(16x4), y(16x8)
  __shared__ float smem[WAVES_PER_BLOCK][256 + 64 + 64 + 64 + 128];
  float* zb  = &smem[wave][0];
  float* h0b = zb + 256;
  float* h1b = h0b + 64;
  float* rb  = h1b + 64;
  float* yb  = rb + 64;

  const int n   = lane & 15;                   // gate/column index (B, C, D layouts)
  const int mA  = lane & 15;                   // sample/row index (A layout)
  const int klo = (lane < 16) ? 0 : 2;         // K pair owned by this half-wave

  // ---- loop-invariant B-matrix fragments: B[k][n] = W[n][k] ----
  v2f eBx0a = { eW0[n * 8 + klo],     eW0[n * 8 + klo + 1] };
  v2f eBx0b = { eW0[n * 8 + 4 + klo], eW0[n * 8 + 5 + klo] };
  v2f eBh0  = { eU0[n * 4 + klo],     eU0[n * 4 + klo + 1] };
  v2f eBx1  = { eW1[n * 4 + klo],     eW1[n * 4 + klo + 1] };
  v2f eBh1  = { eU1[n * 4 + klo],     eU1[n * 4 + klo + 1] };
  v2f dBx0a = { dW0[n * 8 + klo],     dW0[n * 8 + klo + 1] };
  v2f dBx0b = { dW0[n * 8 + 4 + klo], dW0[n * 8 + 5 + klo] };
  v2f dBh0  = { dU0[n * 4 + klo],     dU0[n * 4 + klo + 1] };
  v2f dBx1  = { dW1[n * 4 + klo],     dW1[n * 4 + klo + 1] };
  v2f dBh1  = { dU1[n * 4 + klo],     dU1[n * 4 + klo + 1] };
  v2f Bfc   = { 0.0f, 0.0f };
  float bfc = 0.0f;
  if (n < F_) { Bfc.x = fcW[n * 4 + klo]; Bfc.y = fcW[n * 4 + klo + 1]; bfc = fcb[n]; }
  const float be0 = eb0[n], be1 = eb1[n], bd0 = db0[n], bd1 = db1[n];

  // Elementwise ownership: lane -> pairs (m0,j) and (m1,j)
  const int j  = lane & 3;
  const int m0 = lane >> 2;        // samples 0..7
  const int m1 = 8 + (lane >> 2);  // samples 8..15

  float c0[2] = { 0.0f, 0.0f };    // layer-0 cell state (2 pairs per lane)
  float c1[2] = { 0.0f, 0.0f };    // layer-1 cell state

  // Zero the recurrent state + decoder feedback buffers.
  for (int i = lane; i < 64; i += 32) { h0b[i] = 0.0f; h1b[i] = 0.0f; rb[i] = 0.0f; }
  for (int i = lane; i < 128; i += 32) yb[i] = 0.0f;
  wave_sync_lds();

  // =================== Encoder: 2-layer LSTM over T steps ===================
  for (int t = 0; t < T_; ++t) {
    const float* xp = X + (((size_t)(s0 + mA) * T_ + t) * F_);
    // Prefetch PF_DIST timesteps ahead (512B: future cachelines), WGP scope
    // (locality 3 -> pull into all cache levels). Speculative: safe past end.
    __builtin_prefetch(xp + F_ * PF_DIST, 0, 3);
    v2f ax0; ax0.x = xp[klo];     ax0.y = xp[klo + 1];
    v2f ax1; ax1.x = xp[4 + klo]; ax1.y = xp[5 + klo];
    v2f ah0 = loadA4(h0b, mA, klo);

    AccU z;
#pragma unroll
    for (int v = 0; v < 8; ++v) z.f[v] = be0;  // bias in C accumulator
    z.v = wmma4(ax0, eBx0a, z.v);
    z.v = wmma4(ax1, eBx0b, z.v);
    z.v = wmma4(ah0, eBh0,  z.v);
    store_D(zb, z, lane, n);
    wave_sync_lds();

    lstm_ew(zb, h0b, c0, m0, m1, j);           // layer-0 gates -> h0, c0
    wave_sync_lds();

    v2f ax1h = loadA4(h0b, mA, klo);
    v2f ah1  = loadA4(h1b, mA, klo);
    AccU z1;
#pragma unroll
    for (int v = 0; v < 8; ++v) z1.f[v] = be1;
    z1.v = wmma4(ax1h, eBx1, z1.v);
    z1.v = wmma4(ah1,  eBh1, z1.v);
    store_D(zb, z1, lane, n);
    wave_sync_lds();

    lstm_ew(zb, h1b, c1, m0, m1, j);           // layer-1 gates -> h1, c1
    wave_sync_lds();
  }

  // ====== Decoder: autoregressive loop; carries encoder h/c, x0 = 0 ======
  for (int t = 0; t < T_; ++t) {
    v2f ax0 = loadA8(yb, mA, klo, 0);          // x_t = previous y (or 0)
    v2f ax1 = loadA8(yb, mA, klo, 4);
    v2f ah0 = loadA4(h0b, mA, klo);

    AccU z;
#pragma unroll
    for (int v = 0; v < 8; ++v) z.f[v] = bd0;
    z.v = wmma4(ax0, dBx0a, z.v);
    z.v = wmma4(ax1, dBx0b, z.v);
    z.v = wmma4(ah0, dBh0,  z.v);
    store_D(zb, z, lane, n);
    wave_sync_lds();

    lstm_ew(zb, h0b, c0, m0, m1, j);
    wave_sync_lds();

    v2f ax1h = loadA4(h0b, mA, klo);
    v2f ah1  = loadA4(h1b, mA, klo);
    AccU z1;
#pragma unroll
    for (int v = 0; v < 8; ++v) z1.f[v] = bd1;
    z1.v = wmma4(ax1h, dBx1, z1.v);
    z1.v = wmma4(ah1,  dBh1, z1.v);
    store_D(zb, z1, lane, n);
    wave_sync_lds();

    lstm_ew_relu(zb, h1b, rb, c1, m0, m1, j);  // h1 + relu(h1) for FC
    wave_sync_lds();

    // FC: y = relu(h1) @ fcW.T + fcb  (N columns 8..15 are zero-padded)
    v2f ar = loadA4(rb, mA, klo);
    AccU y;
#pragma unroll
    for (int v = 0; v < 8; ++v) y.f[v] = bfc;
    y.v = wmma4(ar, Bfc, y.v);
    if (n < F_) {
#pragma unroll
      for (int v = 0; v < 8; ++v) {
        const int m = (lane < 16) ? v : (v + 8);
        yb[m * 8 + n] = y.f[v];
      }
    }
    wave_sync_lds();

    // Coalesced, non-temporal store of y to out[:, T-1-t, :] (time-reversed).
    // Output is write-once / never re-read: NT hint keeps it out of near caches.
    {
      const int m    = lane >> 1;
      const int half = lane & 1;
      const v4f yv = *reinterpret_cast<const v4f*>(&yb[m * 8 + half * 4]);
      v4f* op = reinterpret_cast<v4f*>(
          out + (((size_t)(s0 + m) * T_ + (T_ - 1 - t)) * F_ + half * 4));
      __builtin_nontemporal_store(yv, op);
    }
  }
}

extern "C" void kernel_launch(void* const* d_in, const int* in_sizes, int n_in,
                              void* d_out, int out_size, void* d_ws, size_t ws_size,
                              hipStream_t stream) {
  const float* X   = (const float*)d_in[0];
  const float* eW0 = (const float*)d_in[1];
  const float* eU0 = (const float*)d_in[2];
  const float* eb0 = (const float*)d_in[3];
  const float* eW1 = (const float*)d_in[4];
  const float* eU1 = (const float*)d_in[5];
  const float* eb1 = (const float*)d_in[6];
  const float* dW0 = (const float*)d_in[7];
  const float* dU0 = (const float*)d_in[8];
  const float* db0 = (const float*)d_in[9];
  const float* dW1 = (const float*)d_in[10];
  const float* dU1 = (const float*)d_in[11];
  const float* db1 = (const float*)d_in[12];
  const float* fcW = (const float*)d_in[13];
  const float* fcb = (const float*)d_in[14];
  float* out = (float*)d_out;

  const int waves  = B_ / SAMPLES_PER_WAVE;          // 256 waves
  const int blocks = waves / WAVES_PER_BLOCK;        // 32 blocks
  lstm_autoencoder_kernel<<<blocks, 256, 0, stream>>>(
      X, eW0, eU0, eb0, eW1, eU1, eb1,
      dW0, dU0, db0, dW1, dU1, db1, fcW, fcb, out);
}